// GATLayerImp3_50878182589044
// MI455X (gfx1250) — compile-verified
//
#include <hip/hip_runtime.h>
#include <hip/hip_bf16.h>

typedef __attribute__((ext_vector_type(2))) float v2f;
typedef __attribute__((ext_vector_type(8))) float v8f;
typedef __attribute__((address_space(3))) float lds_f;

#define GAT_B   4
#define GAT_N   10000
#define GAT_E   160000
#define GAT_M   (GAT_B * GAT_N)      // 40000 rows
#define GAT_F   256                  // FIN == H*FO
#define GAT_H   8
#define GAT_FO  32
#define NEG_SLOPE 0.2f
#define MASK_FILL -9e15f

#define BPITCH  260                  // dword pitch per W column row: 260 % 64 == 4
                                     // -> conflict-free ds_load_b64 B reads

// ---------------------------------------------------------------- utilities
__global__ void k_zero(float* __restrict__ p, size_t n) {
    size_t i = (size_t)blockIdx.x * blockDim.x + threadIdx.x;
    if (i < n) p[i] = 0.0f;
}

// ---------------------------------------------------------------- GEMM (WMMA f32 + async LDS staging)
// proj[m][n] = sum_k x[m][k] * W[n][k]   (x @ W^T), M=40000, N=256, K=256
// Block = 8 waves = 8 m-tiles x 1 n-group (128 rows x 64 cols).
// The block's W panel (64 cols x 256 K = 64KB) is staged into LDS ONCE via
// GLOBAL_LOAD_ASYNC_TO_LDS_B128 (ASYNCcnt) and shared by all 8 waves,
// then each wave runs the K loop with V_WMMA_F32_16X16X4_F32, A from global
// (contiguous float2 per the f32 16x4 A layout), B from LDS (ds_load_b64).
__global__ __launch_bounds__(256)
void k_gemm_wmma(const float* __restrict__ x, const float* __restrict__ W,
                 float* __restrict__ proj) {
    __shared__ float ldsB[64 * BPITCH];                 // 66,560 B

    const int tid    = threadIdx.x;
    const int lane   = tid & 31;
    const int wave   = tid >> 5;                        // 0..7
    const int mtile  = blockIdx.x * 8 + wave;           // 0..2503 (2500 valid)
    const int ncol_base = blockIdx.y * 64;              // 0,64,128,192

    // ---- async stage: 64 cols x 256 K floats, 16B per lane per issue
    #pragma unroll
    for (int i = 0; i < 16; ++i) {
        const int c   = i * 256 + tid;                  // 0..4095 chunk id
        const int col = c >> 6;                         // 0..63
        const int kk  = (c & 63) * 4;                   // 0..252
        const float* gsrc = W + (size_t)(ncol_base + col) * GAT_F + kk;
        lds_f* lp = (lds_f*)&ldsB[col * BPITCH + kk];
        const unsigned ldst = (unsigned)(unsigned long long)lp;
        asm volatile("global_load_async_to_lds_b128 %0, %1, off"
                     :: "v"(ldst), "v"(gsrc) : "memory");
    }
    asm volatile("s_wait_asynccnt 0" ::: "memory");
    __syncthreads();

    if (mtile * 16 < GAT_M) {                           // wave-uniform guard
        const int kphase = (lane >> 4) * 2;             // 0 or 2
        const float* __restrict__ xrow =
            x + (size_t)(mtile * 16 + (lane & 15)) * GAT_F;

        v8f acc[4];
        acc[0] = (v8f){}; acc[1] = (v8f){}; acc[2] = (v8f){}; acc[3] = (v8f){};

        for (int k = 0; k < GAT_F; k += 4) {
            v2f a = *(const v2f*)(xrow + k + kphase);
            #pragma unroll
            for (int t = 0; t < 4; ++t) {
                // B fragment: lane holds col N = t*16 + (lane&15), K = kphase..+1
                const float* bp =
                    &ldsB[(t * 16 + (lane & 15)) * BPITCH + k + kphase];
                v2f b = *(const v2f*)bp;
                acc[t] = __builtin_amdgcn_wmma_f32_16x16x4_f32(
                    false, a, false, b, (short)0, acc[t], false, false);
            }
        }

        const int rbase = mtile * 16 + 8 * (lane >> 4);
        const int cbase = ncol_base + (lane & 15);
        #pragma unroll
        for (int t = 0; t < 4; ++t) {
            #pragma unroll
            for (int g = 0; g < 8; ++g) {
                proj[(size_t)(rbase + g) * GAT_F + cbase + t * 16] = acc[t][g];
            }
        }
    }
}

// ---------------------------------------------------------------- node scores
// s_src[m][h] = sum_f proj[m][h*32+f] * a_src[h][f]   (same for trg)
__global__ __launch_bounds__(256)
void k_scores(const float* __restrict__ proj,
              const float* __restrict__ a_src, const float* __restrict__ a_trg,
              float* __restrict__ s_src, float* __restrict__ s_trg) {
    const int lane  = threadIdx.x & 31;
    const int node  = (blockIdx.x * blockDim.x + threadIdx.x) >> 5;
    if (node >= GAT_M) return;
    const float* __restrict__ p = proj + (size_t)node * GAT_F;
    #pragma unroll
    for (int h = 0; h < GAT_H; ++h) {
        float v  = p[h * GAT_FO + lane];
        float ss = v * a_src[h * GAT_FO + lane];
        float st = v * a_trg[h * GAT_FO + lane];
        #pragma unroll
        for (int off = 16; off > 0; off >>= 1) {
            ss += __shfl_down(ss, off, 32);
            st += __shfl_down(st, off, 32);
        }
        if (lane == 0) {
            s_src[node * GAT_H + h] = ss;
            s_trg[node * GAT_H + h] = st;
        }
    }
}

// ---------------------------------------------------------------- edge pass 1
// ex[e][h] = exp(mask? lrelu(s_src[src]+s_trg[trg]) : MASK_FILL); den[trg][h] += ex
__global__ __launch_bounds__(256)
void k_edge1(const long long* __restrict__ ei, const float* __restrict__ emask,
             const float* __restrict__ s_src, const float* __restrict__ s_trg,
             float* __restrict__ ex, float* __restrict__ den) {
    const int e = blockIdx.x * blockDim.x + threadIdx.x;   // 0 .. B*E-1
    if (e >= GAT_B * GAT_E) return;
    const int b  = e / GAT_E;
    const int el = e - b * GAT_E;
    const long long* __restrict__ eib = ei + (size_t)b * 2 * GAT_E;
    const long long src = eib[el]         + (long long)b * GAT_N;
    const long long trg = eib[GAT_E + el] + (long long)b * GAT_N;
    const float m = emask[(size_t)b * GAT_E + el];

    const float* __restrict__ ps = s_src + src * GAT_H;
    const float* __restrict__ pt = s_trg + trg * GAT_H;
    #pragma unroll
    for (int h = 0; h < GAT_H; ++h) {
        float sc = ps[h] + pt[h];
        sc = (sc > 0.0f) ? sc : sc * NEG_SLOPE;
        sc = (m > 0.0f) ? sc : MASK_FILL;
        float ev = __expf(sc);                 // exp(MASK_FILL) == 0
        ex[(size_t)e * GAT_H + h] = ev;
        atomicAdd(&den[trg * GAT_H + h], ev);
    }
}

// ---------------------------------------------------------------- edge pass 2
// out[trg][h*32+f] += proj[src][h*32+f] * ex[e][h]/(den[trg][h]+1e-16)
// One wave per edge; lanes 0..7 compute attn, broadcast via shfl.
__global__ __launch_bounds__(256)
void k_edge2(const long long* __restrict__ ei,
             const float* __restrict__ ex, const float* __restrict__ den,
             const float* __restrict__ proj, float* __restrict__ out) {
    const int lane  = threadIdx.x & 31;
    const int e     = (blockIdx.x * blockDim.x + threadIdx.x) >> 5;
    if (e >= GAT_B * GAT_E) return;
    const int b  = e / GAT_E;
    const int el = e - b * GAT_E;
    const long long* __restrict__ eib = ei + (size_t)b * 2 * GAT_E;
    const long long src = eib[el]         + (long long)b * GAT_N;
    const long long trg = eib[GAT_E + el] + (long long)b * GAT_N;

    float attn_l = 0.0f;
    if (lane < GAT_H)
        attn_l = ex[(size_t)e * GAT_H + lane] / (den[trg * GAT_H + lane] + 1e-16f);

    const float* __restrict__ psrc = proj + (size_t)src * GAT_F;
    float* __restrict__ ptrg = out + (size_t)trg * GAT_F;
    #pragma unroll
    for (int h = 0; h < GAT_H; ++h) {
        float a = __shfl(attn_l, h, 32);
        atomicAdd(&ptrg[h * GAT_FO + lane], psrc[h * GAT_FO + lane] * a);
    }
}

// ---------------------------------------------------------------- epilogue
__global__ __launch_bounds__(256)
void k_finish(float* __restrict__ out, const float* __restrict__ bias, size_t n) {
    size_t i = (size_t)blockIdx.x * blockDim.x + threadIdx.x;
    if (i < n) {
        float v = out[i] + bias[i & (GAT_F - 1)];
        out[i] = (v > 0.0f) ? v : (__expf(v) - 1.0f);
    }
}

// ---------------------------------------------------------------- launcher
extern "C" void kernel_launch(void* const* d_in, const int* in_sizes, int n_in,
                              void* d_out, int out_size, void* d_ws, size_t ws_size,
                              hipStream_t stream) {
    const float*     x      = (const float*)d_in[0];
    const long long* ei     = (const long long*)d_in[1];  // int64 edge_index
    const float*     emask  = (const float*)d_in[2];
    const float*     W      = (const float*)d_in[3];
    const float*     a_src  = (const float*)d_in[4];
    const float*     a_trg  = (const float*)d_in[5];
    const float*     bias   = (const float*)d_in[6];
    float*           out    = (float*)d_out;

    // workspace layout (floats)
    float* ws    = (float*)d_ws;
    float* proj  = ws;                               // 40000*256  = 10,240,000
    float* s_src = proj  + (size_t)GAT_M * GAT_F;    // 40000*8
    float* s_trg = s_src + (size_t)GAT_M * GAT_H;
    float* den   = s_trg + (size_t)GAT_M * GAT_H;
    float* ex    = den   + (size_t)GAT_M * GAT_H;    // 640000*8

    const size_t out_n = (size_t)GAT_M * GAT_F;      // 10.24M
    const size_t den_n = (size_t)GAT_M * GAT_H;      // 320K

    // zero accumulators (d_out is the aggregation buffer)
    k_zero<<<(int)((out_n + 255) / 256), 256, 0, stream>>>(out, out_n);
    k_zero<<<(int)((den_n + 255) / 256), 256, 0, stream>>>(den, den_n);

    // GEMM: grid = (ceil(2500/8) m-blocks, 4 n-groups), 8 waves per block
    dim3 ggrid((2500 + 7) / 8, 4, 1);
    k_gemm_wmma<<<ggrid, 256, 0, stream>>>(x, W, proj);

    // node scores: 40000 waves -> 5000 blocks
    k_scores<<<5000, 256, 0, stream>>>(proj, a_src, a_trg, s_src, s_trg);

    // edge pass 1: 640000 threads -> 2500 blocks
    k_edge1<<<2500, 256, 0, stream>>>(ei, emask, s_src, s_trg, ex, den);

    // edge pass 2: 640000 waves -> 80000 blocks
    k_edge2<<<80000, 256, 0, stream>>>(ei, ex, den, proj, out);

    // bias + ELU in place on d_out
    k_finish<<<(int)((out_n + 255) / 256), 256, 0, stream>>>(out, bias, out_n);
}